// Acquisition_Layer_79413945303678
// MI455X (gfx1250) — compile-verified
//
#include <hip/hip_runtime.h>
#include <stdint.h>

#define HW      262144      // 512*512
#define HALF_N  16777216u   // (8*16*512*512)/2

typedef __attribute__((ext_vector_type(2))) float v2f;
typedef __attribute__((ext_vector_type(8))) float v8f;

// Native single-instruction transcendentals (v_log_f32 = log2, v_exp_f32 = exp2)
#define FAST_LOG2(x) __builtin_amdgcn_logf(x)
#define FAST_EXP2(x) __builtin_amdgcn_exp2f(x)

// ---------------- Threefry-2x32 (matches JAX exactly; pure integer) ----------------
__host__ __device__ static inline void tf2x32(uint32_t k0, uint32_t k1,
                                              uint32_t x0, uint32_t x1,
                                              uint32_t& o0, uint32_t& o1)
{
    const uint32_t ks2 = k0 ^ k1 ^ 0x1BD11BDAu;
    x0 += k0; x1 += k1;
#define TF_R(r) { x0 += x1; x1 = (x1 << (r)) | (x1 >> (32 - (r))); x1 ^= x0; }
    TF_R(13) TF_R(15) TF_R(26) TF_R(6)
    x0 += k1;  x1 += ks2 + 1u;
    TF_R(17) TF_R(29) TF_R(16) TF_R(24)
    x0 += ks2; x1 += k0 + 2u;
    TF_R(13) TF_R(15) TF_R(26) TF_R(6)
    x0 += k0;  x1 += k1 + 3u;
    TF_R(17) TF_R(29) TF_R(16) TF_R(24)
    x0 += k1;  x1 += ks2 + 4u;
    TF_R(13) TF_R(15) TF_R(26) TF_R(6)
    x0 += ks2; x1 += k0 + 5u;
#undef TF_R
    o0 = x0; o1 = x1;
}

// ---------------- XLA's f32 erfinv (Giles), w = -log1p(-x^2) ----------------
__device__ __forceinline__ float erfinv_xla(float x)
{
    float w = -log1pf(-x * x);
    float p;
    if (w < 5.0f) {
        w -= 2.5f;
        p = 2.81022636e-08f;
        p = __fmaf_rn(p, w, 3.43273939e-07f);
        p = __fmaf_rn(p, w, -3.5233877e-06f);
        p = __fmaf_rn(p, w, -4.39150654e-06f);
        p = __fmaf_rn(p, w, 0.00021858087f);
        p = __fmaf_rn(p, w, -0.00125372503f);
        p = __fmaf_rn(p, w, -0.00417768164f);
        p = __fmaf_rn(p, w, 0.246640727f);
        p = __fmaf_rn(p, w, 1.50140941f);
    } else {
        w = __fsqrt_rn(w) - 3.0f;
        p = -0.000200214257f;
        p = __fmaf_rn(p, w, 0.000100950558f);
        p = __fmaf_rn(p, w, 0.00134934322f);
        p = __fmaf_rn(p, w, -0.00367342844f);
        p = __fmaf_rn(p, w, 0.00573950773f);
        p = __fmaf_rn(p, w, -0.0076224613f);
        p = __fmaf_rn(p, w, 0.00943887047f);
        p = __fmaf_rn(p, w, 1.00167406f);
        p = __fmaf_rn(p, w, 2.83297682f);
    }
    return p * x;
}

__device__ __forceinline__ float bits01(uint32_t b)
{
    return __uint_as_float((b >> 9) | 0x3F800000u) - 1.0f;   // JAX uniform [0,1)
}

// One acquired-frame sample: clip(base + fw*(-log u)^(-fa) + gw*normal, 0, 1)
__device__ __forceinline__ float sample_one(float base, float fw, float fa, float gw,
                                            uint32_t ubits, uint32_t nbits)
{
    float u = bits01(ubits);
    u = fminf(fmaxf(u, 1e-6f), 1.0f - 1e-6f);
    float neglog = -log1pf(u - 1.0f);                       // accurate -ln(u), u in [0,1)
    float pt = FAST_EXP2(-fa * FAST_LOG2(neglog));          // (-ln u)^(-fa)
    // normal: u' in [lo, 1), lo = nextafter(-1,0); (hi-lo) rounds to 2.0f in f32
    float x = __fmaf_rn(bits01(nbits), 2.0f, -0.99999994f);
    x = fmaxf(x, -0.99999994f);
    float nrm = 1.41421354f * erfinv_xla(x);                // sqrt(2)*erfinv
    float s = base + fw * pt + gw * nrm;
    return fminf(fmaxf(s, 0.0f), 1.0f);
}

// ---------------- per-batch mean of channel 1 via WMMA row-sum (f32, exact FMA) ----
__global__ void bias_reduce_kernel(const float* __restrict__ in, float* __restrict__ bias)
{
    const int b    = blockIdx.x;          // 8 blocks, one per batch
    const int tid  = threadIdx.x;         // 256 threads = 8 waves
    const int lane = tid & 31;
    const int wave = tid >> 5;

    // channel-1 slice: contiguous 262144 floats; wave w owns 32768 of them
    const float* src = in + (size_t)(b * 5 + 1) * HW + (size_t)wave * 32768 + 2 * lane;

    v2f ones; ones.x = 1.0f; ones.y = 1.0f;   // B = 4x16 all-ones -> D rows = row sums of A
    v8f c = {};
    for (int i = 0; i < 512; ++i) {
        v2f a = *(const v2f*)(src + (size_t)i * 64);   // 64 f32 per wave per WMMA
        c = __builtin_amdgcn_wmma_f32_16x16x4_f32(false, a, false, ones,
                                                  (short)0, c, false, false);
    }
    // column N=0 lives in lane0 (M=0..7) and lane16 (M=8..15); sum of those 16 = total
    float s = c[0] + c[1] + c[2] + c[3] + c[4] + c[5] + c[6] + c[7];
    float part = __shfl(s, 0, 32) + __shfl(s, 16, 32);

    __shared__ float lds[8];
    if (lane == 0) lds[wave] = part;
    __syncthreads();
    if (tid == 0) {
        float t = 0.0f;
        for (int w = 0; w < 8; ++w) t += lds[w];       // fixed order: deterministic
        bias[b] = fmaxf(t * (1.0f / (float)HW), 0.0f); // relu(mean)
    }
}

// ---------------- main elementwise kernel: one thread = pixel p of batches b and b+4 ----
__global__ void acquire_main_kernel(const float* __restrict__ in,
                                    const int*   __restrict__ frames,
                                    const float* __restrict__ bias,
                                    float*       __restrict__ out,
                                    uint32_t ku0, uint32_t ku1,
                                    uint32_t kn0, uint32_t kn1)
{
    const int p  = blockIdx.x * blockDim.x + threadIdx.x;  // 0..HW-1
    const int b  = blockIdx.y;                              // 0..3
    const int b4 = b + 4;
    if (p >= HW) return;

    // channel loads (relu applied to fw/fa/gw as in reference)
    const float img_a = in[(b  * 5 + 0) * HW + p];
    const float fw_a  = fmaxf(in[(b  * 5 + 2) * HW + p], 0.0f);
    const float fa_a  = fmaxf(in[(b  * 5 + 3) * HW + p], 0.0f);
    const float gw_a  = fmaxf(in[(b  * 5 + 4) * HW + p], 0.0f);
    const float img_b = in[(b4 * 5 + 0) * HW + p];
    const float fw_b  = fmaxf(in[(b4 * 5 + 2) * HW + p], 0.0f);
    const float fa_b  = fmaxf(in[(b4 * 5 + 3) * HW + p], 0.0f);
    const float gw_b  = fmaxf(in[(b4 * 5 + 4) * HW + p], 0.0f);

    const float base_a = img_a + bias[b];
    const float base_b = img_b + bias[b4];

    // center = clip(base + fw * (1/(fa+1))^fa, 0, 1)
    const float cen_a = fminf(fmaxf(base_a + fw_a * FAST_EXP2(-fa_a * FAST_LOG2(fa_a + 1.0f)), 0.0f), 1.0f);
    const float cen_b = fminf(fmaxf(base_b + fw_b * FAST_EXP2(-fa_b * FAST_LOG2(fa_b + 1.0f)), 0.0f), 1.0f);

    const int na = frames[b];
    const int nb = frames[b4];
    const int nmax = (na > nb) ? na : nb;

    float acc_a = 0.0f, acc_b = 0.0f;
    const uint32_t jbase = (uint32_t)(b * 16) * (uint32_t)HW + (uint32_t)p;

    for (int f = 0; f < nmax; ++f) {
        const uint32_t j = jbase + (uint32_t)f * (uint32_t)HW;
        // JAX counter pairing: out0 -> index j (batch b), out1 -> index j+N/2 (batch b+4)
        uint32_t ua, ub, na_, nb_;
        tf2x32(ku0, ku1, j, j + HALF_N, ua, ub);
        tf2x32(kn0, kn1, j, j + HALF_N, na_, nb_);
        if (f < na) acc_a += sample_one(base_a, fw_a, fa_a, gw_a, ua, na_);
        if (f < nb) acc_b += sample_one(base_b, fw_b, fa_b, gw_b, ub, nb_);
    }

    out[b  * HW + p] = acc_a / (float)na;
    out[b4 * HW + p] = acc_b / (float)nb;
    out[8 * HW + b  * HW + p] = cen_a;
    out[8 * HW + b4 * HW + p] = cen_b;
}

extern "C" void kernel_launch(void* const* d_in, const int* in_sizes, int n_in,
                              void* d_out, int out_size, void* d_ws, size_t ws_size,
                              hipStream_t stream)
{
    const float* in     = (const float*)d_in[0];   // (8,5,512,512) f32
    const int*   frames = (const int*)  d_in[1];   // (8,) i32
    float*       out    = (float*)d_out;           // acquired(8*HW) ++ center(8*HW)
    float*       bias   = (float*)d_ws;            // 8 floats scratch

    // Host-side JAX key derivation: key(42) = (0,42); split -> ku, kn
    // split: threefry over iota(4) with x0=[0,1], x1=[2,3] -> bits[0..3]
    uint32_t bits0, bits1, bits2, bits3;
    tf2x32(0u, 42u, 0u, 2u, bits0, bits2);   // pair (0,2): out0=bits[0], out1=bits[2]
    tf2x32(0u, 42u, 1u, 3u, bits1, bits3);   // pair (1,3): out0=bits[1], out1=bits[3]
    const uint32_t ku0 = bits0, ku1 = bits1; // ku = (bits[0], bits[1])
    const uint32_t kn0 = bits2, kn1 = bits3; // kn = (bits[2], bits[3])

    bias_reduce_kernel<<<dim3(8), dim3(256), 0, stream>>>(in, bias);
    acquire_main_kernel<<<dim3(HW / 256, 4), dim3(256), 0, stream>>>(
        in, frames, bias, out, ku0, ku1, kn0, kn1);
}